// Filter_90142773608790
// MI455X (gfx1250) — compile-verified
//
#include <hip/hip_runtime.h>
#include <cfloat>
#include <math.h>

typedef __attribute__((ext_vector_type(2))) float    v2f;
typedef __attribute__((ext_vector_type(8))) float    v8f;
typedef __attribute__((ext_vector_type(4))) float    v4f;
typedef __attribute__((ext_vector_type(4))) unsigned v4u;

#define N_        32
#define F_        2999
#define T_        400
#define FC_       1199
#define NOISE_LO  999           // band bins [0, 999) are noise
#define NOISE_HI  1400          // band bins [1400, F) are noise
#define PLANE     (F_ * T_)     // 1,199,600 floats per (n,c) plane
#define R1LEN     (NOISE_LO * T_)                 // 399,600 (mult of 4)
#define R2OFF     (NOISE_HI * T_)                 // 560,000
#define NOISE_LEN (R1LEN + (F_ - NOISE_HI) * T_)  // 1,039,200
#define CHUNK     65536
#define CHUNKS_PP 16            // ceil(NOISE_LEN / CHUNK); chunks 0..14 fully in-bounds
#define FULLCH    15

// remap virtual noise index -> float offset within an (n,c) plane
__device__ __forceinline__ int noise_off(int v) {
  return (v < R1LEN) ? v : (v + (R2OFF - R1LEN));
}

// ---------------- Kernel A: per-sample argmax (mid) and windowed sig ----------------
__global__ void mid_sig_kernel(const float* __restrict__ x,
                               int* __restrict__ mid_out, float* __restrict__ sig_out) {
  const int n = blockIdx.x;       // 0..31
  const int tid = threadIdx.x;    // 512 threads
  __shared__ float smag[512];
  __shared__ float sv[512];
  __shared__ int   si[512];

  float mg = 0.f;
  float m  = -FLT_MAX;
  if (tid < T_) {
    const long b0 = ((long)(n * 2 + 0) * F_ + FC_) * T_ + tid;
    const long b1 = ((long)(n * 2 + 1) * F_ + FC_) * T_ + tid;
    mg = fabsf(x[b0]) + fabsf(x[b1]);
    m = mg;
  }
  smag[tid] = mg;
  sv[tid] = m;
  si[tid] = tid;
  __syncthreads();

  // argmax, first-index tie-break (matches jnp.argmax)
  for (int s = 256; s > 0; s >>= 1) {
    if (tid < s) {
      const float v2 = sv[tid + s];
      const int   i2 = si[tid + s];
      if (v2 > sv[tid] || (v2 == sv[tid] && i2 < si[tid])) { sv[tid] = v2; si[tid] = i2; }
    }
    __syncthreads();
  }
  const int mid = si[0];
  __syncthreads();

  // sig = sum of mag over t in [mid-20, mid+20) clipped to [0, T)
  sv[tid] = (tid < T_ && tid >= mid - 20 && tid < mid + 20) ? smag[tid] : 0.f;
  __syncthreads();
  for (int s = 256; s > 0; s >>= 1) {
    if (tid < s) sv[tid] += sv[tid + s];
    __syncthreads();
  }
  if (tid == 0) { mid_out[n] = mid; sig_out[n] = sv[0]; }
}

// ---------------- Kernel B: noise |x| partial sums via WMMA f32 16x16x4 ----------------
// B-matrix = ones: each wmma folds 64 floats/wave into the f32 accumulator.
// sum(all 256 accumulator elems) == 16 * sum(inputs fed).
__global__ void noise_partial_kernel(const float* __restrict__ x, float* __restrict__ partial) {
  const int b = blockIdx.x;                 // 1024 blocks = 32n * 2c * 16 chunks
  const int nc = b >> 4;
  const int chunk = b & 15;
  const float* plane = x + (long)nc * PLANE;

  const int v0 = chunk * CHUNK + (int)threadIdx.x * 4;   // 4 floats/lane/sub-iter
  v8f acc = {};
  v2f ones; ones.x = 1.0f; ones.y = 1.0f;

  if (chunk < FULLCH) {
    // hot path: no bounds checks, 4 b128 loads in flight per wave, then 8 wmmas
    for (int i = 0; i < CHUNK / 4096; ++i) {             // 16 outer iterations
      v4f p[4];
      #pragma unroll
      for (int j = 0; j < 4; ++j) {
        const int v = v0 + (i * 4 + j) * 1024;           // 1024 floats/block/sub-iter
        p[j] = __builtin_nontemporal_load(
            reinterpret_cast<const v4f*>(plane + noise_off(v)));
      }
      #pragma unroll
      for (int j = 0; j < 4; ++j) {
        v2f a0; a0.x = fabsf(p[j].x); a0.y = fabsf(p[j].y);
        v2f a1; a1.x = fabsf(p[j].z); a1.y = fabsf(p[j].w);
        acc = __builtin_amdgcn_wmma_f32_16x16x4_f32(false, a0, false, ones,
                                                    (short)0, acc, false, false);
        acc = __builtin_amdgcn_wmma_f32_16x16x4_f32(false, a1, false, ones,
                                                    (short)0, acc, false, false);
      }
    }
  } else {
    // tail chunk: predicated loads (zero-fill), wmma stays EXEC-all-ones
    for (int i = 0; i < CHUNK / 1024; ++i) {
      const int v = v0 + i * 1024;
      v4f p; p.x = 0.f; p.y = 0.f; p.z = 0.f; p.w = 0.f;
      if (v < NOISE_LEN) {                               // quads never straddle seam/end
        p = __builtin_nontemporal_load(
            reinterpret_cast<const v4f*>(plane + noise_off(v)));
      }
      v2f a0; a0.x = fabsf(p.x); a0.y = fabsf(p.y);
      v2f a1; a1.x = fabsf(p.z); a1.y = fabsf(p.w);
      acc = __builtin_amdgcn_wmma_f32_16x16x4_f32(false, a0, false, ones,
                                                  (short)0, acc, false, false);
      acc = __builtin_amdgcn_wmma_f32_16x16x4_f32(false, a1, false, ones,
                                                  (short)0, acc, false, false);
    }
  }

  float s = acc[0] + acc[1] + acc[2] + acc[3] + acc[4] + acc[5] + acc[6] + acc[7];
  __shared__ float red[256];
  red[threadIdx.x] = s;
  __syncthreads();
  for (int st = 128; st > 0; st >>= 1) {
    if ((int)threadIdx.x < st) red[threadIdx.x] += red[threadIdx.x + st];
    __syncthreads();
  }
  if (threadIdx.x == 0) partial[b] = red[0] * (1.0f / 16.0f);
}

// ---------------- Kernel C: snr/hb + factored keep mask bit-tables ----------------
__global__ void mask_kernel(const int* __restrict__ mid_in, const float* __restrict__ sig_in,
                            const float* __restrict__ partial,
                            unsigned* __restrict__ tbits, unsigned* __restrict__ fbits) {
  __shared__ int s_mid[N_];
  __shared__ int s_hb[N_];
  const int tid = threadIdx.x;

  if (tid < N_) {
    float ns = 0.f;
    for (int j = 0; j < 2 * CHUNKS_PP; ++j) ns += partial[tid * (2 * CHUNKS_PP) + j];
    const float noise = ns * (1.0f / (float)NOISE_LEN);
    const float sig = sig_in[tid];
    const float d = sig - noise;
    const float snr = 10.0f * log10f((d * d) / (noise * noise));
    int hb = (int)truncf(6.0f * (snr - 48.0f) + 27.0f);
    if (hb < 8) hb = 8;
    s_hb[tid] = hb;
    s_mid[tid] = mid_in[tid];
  }
  __syncthreads();

  for (int t = tid; t < T_; t += (int)blockDim.x) {
    unsigned m = 0;
    #pragma unroll
    for (int n = 0; n < N_; ++n) {
      const int mid = s_mid[n];
      if (t >= mid - 8 && t < mid + 8) m |= (1u << n);
    }
    tbits[t] = m;
  }
  for (int f = tid; f < F_; f += (int)blockDim.x) {
    unsigned m = 0;
    #pragma unroll
    for (int n = 0; n < N_; ++n) {
      const int hb = s_hb[n];
      if (f >= FC_ - hb && f < FC_ + hb) m |= (1u << n);
    }
    fbits[f] = m;
  }
}

// ---------------- Kernel D: out = x * keep[f,t], 614 MB NT stream ----------------
__global__ void apply_kernel(const float* __restrict__ x, float* __restrict__ out,
                             const unsigned* __restrict__ tbits,
                             const unsigned* __restrict__ fbits, int total4) {
  const int gid = blockIdx.x * blockDim.x + threadIdx.x;
  if (gid >= total4) return;
  const int base = gid * 4;           // T_%4==0 -> float4 stays within one (f) row
  const int t = base % T_;
  const int row = base / T_;
  const int f = row % F_;

  const v4f v = __builtin_nontemporal_load(reinterpret_cast<const v4f*>(x) + gid);
  const unsigned fb = fbits[f];
  const v4u tb = *reinterpret_cast<const v4u*>(tbits + t);
  v4f o;
  o.x = (fb & tb.x) ? v.x : 0.f;
  o.y = (fb & tb.y) ? v.y : 0.f;
  o.z = (fb & tb.z) ? v.z : 0.f;
  o.w = (fb & tb.w) ? v.w : 0.f;
  __builtin_nontemporal_store(o, reinterpret_cast<v4f*>(out) + gid);
}

extern "C" void kernel_launch(void* const* d_in, const int* in_sizes, int n_in,
                              void* d_out, int out_size, void* d_ws, size_t ws_size,
                              hipStream_t stream) {
  const float* x = (const float*)d_in[0];
  float* out = (float*)d_out;
  char* ws = (char*)d_ws;

  // workspace layout (< 18 KB, 16B-aligned chunks)
  int*      mid     = (int*)(ws + 0);            // 32 ints
  float*    sig     = (float*)(ws + 128);        // 32 floats
  float*    partial = (float*)(ws + 256);        // 1024 floats (ends 4352)
  unsigned* tbits   = (unsigned*)(ws + 4352);    // 400 u32 (ends 5952)
  unsigned* fbits   = (unsigned*)(ws + 5952);    // 2999 u32 (ends 17948)

  mid_sig_kernel<<<N_, 512, 0, stream>>>(x, mid, sig);
  noise_partial_kernel<<<N_ * 2 * CHUNKS_PP, 256, 0, stream>>>(x, partial);
  mask_kernel<<<1, 256, 0, stream>>>(mid, sig, partial, tbits, fbits);

  const int total4 = out_size / 4;
  apply_kernel<<<(total4 + 255) / 256, 256, 0, stream>>>(x, out, tbits, fbits, total4);
}